// LinearAttention_74285754352054
// MI455X (gfx1250) — compile-verified
//
#include <hip/hip_runtime.h>
#include <hip/hip_bf16.h>

// ---------------------------------------------------------------------------
// LinearAttention (b=4, c=128, n=32768, heads=4, dim_head=32), fp32 end-to-end.
// Matrix math on V_WMMA_F32_16X16X4_F32 (CDNA5 wave32 WMMA, full fp32).
// LDS tile staging via GLOBAL_LOAD_ASYNC_TO_LDS_B128 (ASYNCcnt) when available.
// ---------------------------------------------------------------------------

typedef float v2f __attribute__((ext_vector_type(2)));
typedef float v8f __attribute__((ext_vector_type(8)));
typedef int la_v4i __attribute__((vector_size(16)));   // matches builtin's b128 operand

#define N_SPATIAL 32768
#define C_CH      128
#define BATCH     4
#define HIDDEN    128
#define QKV_ROWS  384
#define PER_SAMPLE ((size_t)C_CH * N_SPATIAL)   // 4194304 = 2^22
#define STAT_CHUNKS 1024
#define EPS 1e-5f
#define SCALE 0.17677669529663687f              // 32^-0.5

// workspace layout (float offsets)
#define OFF_SP1  ((size_t)0)                          // 4*1024*2
#define OFF_ST1  (OFF_SP1 + 8192)                     // 4*2
#define OFF_W1F  (OFF_ST1 + 8)                        // 4*384*128
#define OFF_B1F  (OFF_W1F + 196608)                   // 4*384
#define OFF_QKV  (OFF_B1F + 1536)                     // 4*384*32768
#define OFF_CTXP (OFF_QKV + 50331648)                 // 16*64*1024
#define OFF_CTX  (OFF_CTXP + 1048576)                 // 16*32*32
#define OFF_MC   (OFF_CTX + 16384)                    // 4*128*128
#define OFF_SP2  (OFF_MC + 65536)                     // 4*1024*2
#define OFF_ST2  (OFF_SP2 + 8192)                     // 4*2

#if defined(__has_builtin)
#if __has_builtin(__builtin_amdgcn_global_load_async_to_lds_b128) && \
    __has_builtin(__builtin_amdgcn_s_wait_asynccnt)
#define LA_ASYNC_LDS 1
#endif
#endif

__device__ __forceinline__ v8f wmma_f32(v2f a, v2f b, v8f c) {
  // D = A(16x4) * B(4x16) + C(16x16), all fp32, wave32.
  return __builtin_amdgcn_wmma_f32_16x16x4_f32(
      /*neg_a=*/false, a, /*neg_b=*/false, b,
      /*c_mod=*/(short)0, c, /*reuse_a=*/false, /*reuse_b=*/false);
}

// Stage a 128x32 fp32 tile (rows stride N_SPATIAL) into LDS [row][col].
// 256 threads; async direct-to-LDS path bypasses the register file.
__device__ __forceinline__ void la_stage_tile(const float* __restrict__ src,
                                              float* dst_lds, int tid) {
#ifdef LA_ASYNC_LDS
  #pragma unroll
  for (int i = 0; i < 4; ++i) {
    int id = tid + i * 256;                  // 0..1023 float4 slots
    int c = id >> 3, c4 = id & 7;
    __builtin_amdgcn_global_load_async_to_lds_b128(
        (la_v4i __attribute__((address_space(1)))*)(src + (size_t)c * N_SPATIAL + c4 * 4),
        (la_v4i __attribute__((address_space(3)))*)(dst_lds + c * 32 + c4 * 4),
        /*imm_offset=*/0, /*cpol=*/0);
  }
  __builtin_amdgcn_s_wait_asynccnt(0);       // this wave's LDS writes visible
#else
  #pragma unroll
  for (int i = 0; i < 4; ++i) {
    int id = tid + i * 256;
    int c = id >> 3, c4 = id & 7;
    *(float4*)(dst_lds + c * 32 + c4 * 4) =
        *(const float4*)(src + (size_t)c * N_SPATIAL + c4 * 4);
  }
#endif
}

// ---------------------------------------------------------------------------
// GroupNorm statistics: stage 1 (per-chunk partial sum / sumsq)
// grid (STAT_CHUNKS, BATCH), block 256
// ---------------------------------------------------------------------------
__global__ void __launch_bounds__(256)
la_stats_partial(const float* __restrict__ t, float* __restrict__ part) {
  const int s = blockIdx.y;
  const int chunk = blockIdx.x;
  const size_t cnt = PER_SAMPLE / STAT_CHUNKS;   // 4096
  const float* p = t + (size_t)s * PER_SAMPLE + (size_t)chunk * cnt;
  float sum = 0.f, sq = 0.f;
  for (size_t i = threadIdx.x; i < cnt; i += 256) {
    float v = p[i];
    sum += v; sq += v * v;
  }
  __shared__ float smA[256], smB[256];
  smA[threadIdx.x] = sum; smB[threadIdx.x] = sq;
  __syncthreads();
  for (int st = 128; st > 0; st >>= 1) {
    if ((int)threadIdx.x < st) {
      smA[threadIdx.x] += smA[threadIdx.x + st];
      smB[threadIdx.x] += smB[threadIdx.x + st];
    }
    __syncthreads();
  }
  if (threadIdx.x == 0) {
    part[(size_t)(s * STAT_CHUNKS + chunk) * 2 + 0] = smA[0];
    part[(size_t)(s * STAT_CHUNKS + chunk) * 2 + 1] = smB[0];
  }
}

// stage 2: reduce partials -> (mu, rstd) per sample. grid BATCH, block 256
__global__ void __launch_bounds__(256)
la_stats_final(const float* __restrict__ part, float* __restrict__ stats) {
  const int s = blockIdx.x;
  float sum = 0.f, sq = 0.f;
  for (int i = threadIdx.x; i < STAT_CHUNKS; i += 256) {
    sum += part[(size_t)(s * STAT_CHUNKS + i) * 2 + 0];
    sq  += part[(size_t)(s * STAT_CHUNKS + i) * 2 + 1];
  }
  __shared__ float smA[256], smB[256];
  smA[threadIdx.x] = sum; smB[threadIdx.x] = sq;
  __syncthreads();
  for (int st = 128; st > 0; st >>= 1) {
    if ((int)threadIdx.x < st) {
      smA[threadIdx.x] += smA[threadIdx.x + st];
      smB[threadIdx.x] += smB[threadIdx.x + st];
    }
    __syncthreads();
  }
  if (threadIdx.x == 0) {
    const float inv_n = 1.f / (float)PER_SAMPLE;
    float mu = smA[0] * inv_n;
    float var = smB[0] * inv_n - mu * mu;
    stats[s * 2 + 0] = mu;
    stats[s * 2 + 1] = rsqrtf(var + EPS);
  }
}

// ---------------------------------------------------------------------------
// Fold GroupNorm-1 into qkv weights (per sample).
// w1f[s][o][c] = w_qkv[o][c]*gamma1[c]*rstd_s
// b1f[s][o]    = sum_c w_qkv[o][c]*(beta1[c] - mu_s*rstd_s*gamma1[c])
// grid BATCH*QKV_ROWS, block 128
// ---------------------------------------------------------------------------
__global__ void __launch_bounds__(128)
la_fold_w1(const float* __restrict__ w_qkv, const float* __restrict__ gamma1,
           const float* __restrict__ beta1, const float* __restrict__ stats,
           float* __restrict__ w1f, float* __restrict__ b1f) {
  const int blk = blockIdx.x;              // s*384 + o
  const int s = blk / QKV_ROWS;
  const int o = blk % QKV_ROWS;
  const int c = threadIdx.x;
  const float mu = stats[s * 2], rstd = stats[s * 2 + 1];
  const float wv = w_qkv[o * C_CH + c];
  const float g = gamma1[c];
  w1f[((size_t)s * QKV_ROWS + o) * C_CH + c] = wv * g * rstd;
  float bterm = wv * (beta1[c] - mu * rstd * g);
  __shared__ float sm[128];
  sm[c] = bterm;
  __syncthreads();
  for (int st = 64; st > 0; st >>= 1) {
    if (c < st) sm[c] += sm[c + st];
    __syncthreads();
  }
  if (c == 0) b1f[s * QKV_ROWS + o] = sm[0];
}

// ---------------------------------------------------------------------------
// qkv GEMM: qkv[s] = w1f[s](384x128) @ x[s](128xN) + b1f[s]
// grid (N/32, BATCH), block 256 (8 waves). x tile 128x32 staged in LDS.
// Wave w owns M-tiles 3w..3w+2 (48 rows) x both 16-col N-subtiles.
// ---------------------------------------------------------------------------
__global__ void __launch_bounds__(256)
la_qkv_gemm(const float* __restrict__ x, const float* __restrict__ w1f,
            const float* __restrict__ b1f, float* __restrict__ qkv) {
  __shared__ float sB[C_CH * 32];          // [c][col], 16 KB
  const int s = blockIdx.y;
  const int n0 = blockIdx.x * 32;
  const int tid = threadIdx.x;

  la_stage_tile(x + (size_t)s * PER_SAMPLE + n0, sB, tid);
  __syncthreads();

  const int lane = tid & 31;
  const int wave = tid >> 5;
  const int mrow = lane & 15;
  const int koff = (lane >> 4) << 1;       // 0 or 2
  const float* Wb = w1f + (size_t)s * QKV_ROWS * C_CH;

  v8f acc[3][2] = {};
  for (int kk = 0; kk < C_CH; kk += 4) {
    v2f b0, b1;
    b0.x = sB[(kk + koff) * 32 + mrow];
    b0.y = sB[(kk + koff + 1) * 32 + mrow];
    b1.x = sB[(kk + koff) * 32 + 16 + mrow];
    b1.y = sB[(kk + koff + 1) * 32 + 16 + mrow];
    #pragma unroll
    for (int mt = 0; mt < 3; ++mt) {
      int row = (wave * 3 + mt) * 16 + mrow;
      v2f a = *(const v2f*)(Wb + (size_t)row * C_CH + kk + koff);
      acc[mt][0] = wmma_f32(a, b0, acc[mt][0]);
      acc[mt][1] = wmma_f32(a, b1, acc[mt][1]);
    }
  }

  const float* bb = b1f + s * QKV_ROWS;
  #pragma unroll
  for (int mt = 0; mt < 3; ++mt) {
    int mbase = (wave * 3 + mt) * 16;
    #pragma unroll
    for (int r = 0; r < 8; ++r) {
      int row = mbase + ((lane < 16) ? r : r + 8);
      float bias = bb[row];
      size_t orow = ((size_t)s * QKV_ROWS + row) * N_SPATIAL + n0;
      qkv[orow + mrow]      = acc[mt][0][r] + bias;
      qkv[orow + 16 + mrow] = acc[mt][1][r] + bias;
    }
  }
}

// ---------------------------------------------------------------------------
// k softmax over n (in place on qkv rows 128..255 of each sample)
// grid 512 (= BATCH*HIDDEN rows), block 256
// ---------------------------------------------------------------------------
__global__ void __launch_bounds__(256)
la_k_softmax(float* __restrict__ qkv) {
  const int row = blockIdx.x;              // 0..511
  const int s = row >> 7, r = row & 127;
  float* p = qkv + ((size_t)s * QKV_ROWS + HIDDEN + r) * N_SPATIAL;
  const int tid = threadIdx.x;
  __shared__ float sm[256];

  float m = -3.4e38f;
  for (int i = tid; i < N_SPATIAL; i += 256) m = fmaxf(m, p[i]);
  sm[tid] = m; __syncthreads();
  for (int st = 128; st > 0; st >>= 1) {
    if (tid < st) sm[tid] = fmaxf(sm[tid], sm[tid + st]);
    __syncthreads();
  }
  m = sm[0]; __syncthreads();

  float sum = 0.f;
  for (int i = tid; i < N_SPATIAL; i += 256) sum += __expf(p[i] - m);
  sm[tid] = sum; __syncthreads();
  for (int st = 128; st > 0; st >>= 1) {
    if (tid < st) sm[tid] += sm[tid + st];
    __syncthreads();
  }
  float inv = 1.f / sm[0]; __syncthreads();

  for (int i = tid; i < N_SPATIAL; i += 256) p[i] = __expf(p[i] - m) * inv;
}

// ---------------------------------------------------------------------------
// context partials: ctx[bh][d][e] = sum_n k[d,n]*v[e,n], deterministic split-K.
// grid (64 chunks, 16 bh), block 32 (one wave). K-chunk = 512.
// ---------------------------------------------------------------------------
__global__ void __launch_bounds__(32)
la_ctx_partial(const float* __restrict__ qkv, float* __restrict__ ctxp) {
  const int bh = blockIdx.y;
  const int chunk = blockIdx.x;
  const int s = bh >> 2, h = bh & 3;
  const int lane = threadIdx.x & 31;
  const int mrow = lane & 15;
  const int koff = (lane >> 4) << 1;
  const float* kb = qkv + ((size_t)s * QKV_ROWS + HIDDEN + h * 32) * N_SPATIAL;
  const float* vb = qkv + ((size_t)s * QKV_ROWS + 2 * HIDDEN + h * 32) * N_SPATIAL;
  const int nbase = chunk * 512;

  v8f acc[2][2] = {};
  for (int kk = 0; kk < 512; kk += 4) {
    int nn = nbase + kk + koff;
    __builtin_prefetch(kb + (size_t)mrow * N_SPATIAL + nn + 64, 0, 0);
    v2f a0 = *(const v2f*)(kb + (size_t)mrow * N_SPATIAL + nn);
    v2f a1 = *(const v2f*)(kb + (size_t)(16 + mrow) * N_SPATIAL + nn);
    v2f b0 = *(const v2f*)(vb + (size_t)mrow * N_SPATIAL + nn);
    v2f b1 = *(const v2f*)(vb + (size_t)(16 + mrow) * N_SPATIAL + nn);
    acc[0][0] = wmma_f32(a0, b0, acc[0][0]);
    acc[0][1] = wmma_f32(a0, b1, acc[0][1]);
    acc[1][0] = wmma_f32(a1, b0, acc[1][0]);
    acc[1][1] = wmma_f32(a1, b1, acc[1][1]);
  }
  float* p = ctxp + (size_t)(bh * 64 + chunk) * 1024;
  #pragma unroll
  for (int mt = 0; mt < 2; ++mt)
    #pragma unroll
    for (int nt = 0; nt < 2; ++nt) {
      int tile = mt * 2 + nt;
      #pragma unroll
      for (int r = 0; r < 8; ++r) p[tile * 256 + lane * 8 + r] = acc[mt][nt][r];
    }
}

// reduce 64 chunk-partials, decode WMMA C layout -> ctx[bh][d][e]. grid 16 x 256
__global__ void __launch_bounds__(256)
la_ctx_reduce(const float* __restrict__ ctxp, float* __restrict__ ctx) {
  const int bh = blockIdx.x;
  for (int t = threadIdx.x; t < 1024; t += 256) {
    float v = 0.f;
    for (int ch = 0; ch < 64; ++ch) v += ctxp[(size_t)(bh * 64 + ch) * 1024 + t];
    int tile = t >> 8, lane = (t >> 3) & 31, r = t & 7;
    int mt = tile >> 1, nt = tile & 1;
    int d = mt * 16 + ((lane < 16) ? r : r + 8);
    int e = nt * 16 + (lane & 15);
    ctx[(size_t)bh * 1024 + d * 32 + e] = v;
  }
}

// ---------------------------------------------------------------------------
// Fold w_out with context: mc[s][o][h*32+d] = sum_e w_out[o][h*32+e]*ctx[s,h,d,e]
// grid 256 x 256 (65536 outputs)
// ---------------------------------------------------------------------------
__global__ void __launch_bounds__(256)
la_combine(const float* __restrict__ w_out, const float* __restrict__ ctx,
           float* __restrict__ mc) {
  int idx = blockIdx.x * 256 + threadIdx.x;       // s*16384 + o*128 + hd
  int s = idx >> 14;
  int rem = idx & 16383;
  int o = rem >> 7;
  int hd = rem & 127;
  int h = hd >> 5, d = hd & 31;
  const float* wr = w_out + o * HIDDEN + h * 32;
  const float* cr = ctx + (size_t)(s * 4 + h) * 1024 + d * 32;
  float acc = 0.f;
  #pragma unroll
  for (int e = 0; e < 32; ++e) acc += wr[e] * cr[e];
  mc[idx] = acc;
}

// ---------------------------------------------------------------------------
// Output GEMM with fused q-softmax (over d within head, * SCALE) and bias:
// y[s] = mc[s](128x128) @ softmax_d(q[s])(128xN) + b_out
// grid (N/32, BATCH), block 256 (8 waves). Wave w owns M-tile w x 2 N-subtiles.
// ---------------------------------------------------------------------------
__global__ void __launch_bounds__(256)
la_out_gemm(const float* __restrict__ qkv, const float* __restrict__ mc,
            const float* __restrict__ b_out, float* __restrict__ y) {
  __shared__ float sQ[HIDDEN * 32];
  const int s = blockIdx.y;
  const int n0 = blockIdx.x * 32;
  const int tid = threadIdx.x;

  la_stage_tile(qkv + (size_t)s * QKV_ROWS * N_SPATIAL + n0, sQ, tid);  // q rows 0..127
  __syncthreads();

  if (tid < 128) {                       // per (col, head) softmax over 32 d's
    int col = tid & 31, h = tid >> 5;
    float m = -3.4e38f;
    for (int d = 0; d < 32; ++d) m = fmaxf(m, sQ[(h * 32 + d) * 32 + col]);
    float sum = 0.f;
    for (int d = 0; d < 32; ++d) sum += __expf(sQ[(h * 32 + d) * 32 + col] - m);
    float inv = SCALE / sum;
    for (int d = 0; d < 32; ++d) {
      int ix = (h * 32 + d) * 32 + col;
      sQ[ix] = __expf(sQ[ix] - m) * inv;
    }
  }
  __syncthreads();

  const int lane = tid & 31;
  const int wave = tid >> 5;
  const int mrow = lane & 15;
  const int koff = (lane >> 4) << 1;
  const float* A = mc + (size_t)s * HIDDEN * HIDDEN;

  v8f acc0 = {}, acc1 = {};
  for (int kk = 0; kk < HIDDEN; kk += 4) {
    v2f a = *(const v2f*)(A + (size_t)(wave * 16 + mrow) * HIDDEN + kk + koff);
    v2f b0, b1;
    b0.x = sQ[(kk + koff) * 32 + mrow];
    b0.y = sQ[(kk + koff + 1) * 32 + mrow];
    b1.x = sQ[(kk + koff) * 32 + 16 + mrow];
    b1.y = sQ[(kk + koff + 1) * 32 + 16 + mrow];
    acc0 = wmma_f32(a, b0, acc0);
    acc1 = wmma_f32(a, b1, acc1);
  }

  #pragma unroll
  for (int r = 0; r < 8; ++r) {
    int row = wave * 16 + ((lane < 16) ? r : r + 8);
    float bias = b_out[row];
    size_t orow = ((size_t)s * HIDDEN + row) * N_SPATIAL + n0;
    y[orow + mrow]      = acc0[r] + bias;
    y[orow + 16 + mrow] = acc1[r] + bias;
  }
}

// ---------------------------------------------------------------------------
// GroupNorm-2 normalize in place on y. grid 16384 x 256, 4 elems/thread.
// ---------------------------------------------------------------------------
__global__ void __launch_bounds__(256)
la_norm2(float* __restrict__ y, const float* __restrict__ gamma2,
         const float* __restrict__ beta2, const float* __restrict__ stats2) {
  size_t idx = (size_t)blockIdx.x * 256 + threadIdx.x;
  const size_t total = (size_t)BATCH * PER_SAMPLE;
  const size_t stride = (size_t)gridDim.x * 256;
  for (size_t i = idx; i < total; i += stride) {
    int s = (int)(i >> 22);
    int c = (int)((i >> 15) & 127);
    float mu = stats2[s * 2], rstd = stats2[s * 2 + 1];
    y[i] = (y[i] - mu) * rstd * gamma2[c] + beta2[c];
  }
}

// ---------------------------------------------------------------------------
extern "C" void kernel_launch(void* const* d_in, const int* in_sizes, int n_in,
                              void* d_out, int out_size, void* d_ws, size_t ws_size,
                              hipStream_t stream) {
  const float* x      = (const float*)d_in[0];
  const float* gamma1 = (const float*)d_in[1];
  const float* beta1  = (const float*)d_in[2];
  const float* w_qkv  = (const float*)d_in[3];
  const float* w_out  = (const float*)d_in[4];
  const float* b_out  = (const float*)d_in[5];
  const float* gamma2 = (const float*)d_in[6];
  const float* beta2  = (const float*)d_in[7];
  float* y = (float*)d_out;

  float* ws   = (float*)d_ws;
  float* sp1  = ws + OFF_SP1;
  float* st1  = ws + OFF_ST1;
  float* w1f  = ws + OFF_W1F;
  float* b1f  = ws + OFF_B1F;
  float* qkv  = ws + OFF_QKV;
  float* ctxp = ws + OFF_CTXP;
  float* ctx  = ws + OFF_CTX;
  float* mc   = ws + OFF_MC;
  float* sp2  = ws + OFF_SP2;
  float* st2  = ws + OFF_ST2;

  const dim3 blk256(256);
  // 1) GroupNorm-1 statistics
  la_stats_partial<<<dim3(STAT_CHUNKS, BATCH), blk256, 0, stream>>>(x, sp1);
  la_stats_final<<<BATCH, blk256, 0, stream>>>(sp1, st1);
  // 2) fold norm into qkv weights
  la_fold_w1<<<BATCH * QKV_ROWS, 128, 0, stream>>>(w_qkv, gamma1, beta1, st1, w1f, b1f);
  // 3) qkv GEMM (WMMA f32, async LDS staging)
  la_qkv_gemm<<<dim3(N_SPATIAL / 32, BATCH), blk256, 0, stream>>>(x, w1f, b1f, qkv);
  // 4) k softmax over n (in place)
  la_k_softmax<<<BATCH * HIDDEN, blk256, 0, stream>>>(qkv);
  // 5) context = k v^T (WMMA f32, deterministic split-K)
  la_ctx_partial<<<dim3(64, 16), 32, 0, stream>>>(qkv, ctxp);
  la_ctx_reduce<<<16, blk256, 0, stream>>>(ctxp, ctx);
  // 6) fold w_out with context
  la_combine<<<(BATCH * HIDDEN * HIDDEN) / 256, blk256, 0, stream>>>(w_out, ctx, mc);
  // 7) output GEMM with fused q-softmax + bias (WMMA f32, async LDS staging)
  la_out_gemm<<<dim3(N_SPATIAL / 32, BATCH), blk256, 0, stream>>>(qkv, mc, b_out, y);
  // 8) GroupNorm-2
  la_stats_partial<<<dim3(STAT_CHUNKS, BATCH), blk256, 0, stream>>>(y, sp2);
  la_stats_final<<<BATCH, blk256, 0, stream>>>(sp2, st2);
  la_norm2<<<16384, blk256, 0, stream>>>(y, gamma2, beta2, st2);
}